// GCN_42889543418404
// MI455X (gfx1250) — compile-verified
//
#include <hip/hip_runtime.h>
#include <hip/hip_bf16.h>

typedef __attribute__((ext_vector_type(2))) float v2f;
typedef __attribute__((ext_vector_type(8))) float v8f;

#define C_HID 256

// ------------------------------------------------------------------
// Degree computation (exact integer counts, then rsqrt)
// ------------------------------------------------------------------
__global__ void k_deg_init(int* __restrict__ degi, int n) {
  int i = blockIdx.x * blockDim.x + threadIdx.x;
  if (i < n) degi[i] = 1;  // self loop
}

__global__ void k_deg_count(const int* __restrict__ dst, int* __restrict__ degi, int E) {
  int e = blockIdx.x * blockDim.x + threadIdx.x;
  if (e < E) atomicAdd(&degi[dst[e]], 1);
}

__global__ void k_dinv(const int* __restrict__ degi, float* __restrict__ dinv, int n) {
  int i = blockIdx.x * blockDim.x + threadIdx.x;
  if (i < n) dinv[i] = rsqrtf((float)degi[i]);
}

// ------------------------------------------------------------------
// Zero a float buffer
// ------------------------------------------------------------------
__global__ void k_zero(float* __restrict__ p, long long n) {
  long long i = (long long)blockIdx.x * blockDim.x + threadIdx.x;
  if (i < n) p[i] = 0.0f;
}

// ------------------------------------------------------------------
// H = X @ W using V_WMMA_F32_16X16X4_F32 (one wave per 16x16 tile)
//   A layout: lanes 0-15 -> (M=lane, K=k..k+1 in v0,v1); lanes 16-31 -> K=k+2,k+3
//   B layout: lanes 0-15 -> (N=lane, K=k..k+1);          lanes 16-31 -> K=k+2,k+3
//   D layout: vgpr r, lanes 0-15 -> M=r, N=lane; lanes 16-31 -> M=r+8
// ------------------------------------------------------------------
__global__ __launch_bounds__(256) void k_gemm_wmma(
    const float* __restrict__ X, const float* __restrict__ W,
    float* __restrict__ Hout, int M, int K, int N) {
  const int wpb  = blockDim.x >> 5;
  const int wave = blockIdx.x * wpb + ((int)threadIdx.x >> 5);
  const int lane = threadIdx.x & 31;
  const int ntiles = N >> 4;
  const int mt = wave / ntiles;
  const int nt = wave - mt * ntiles;
  const int m0 = mt << 4;
  if (m0 >= M) return;
  const int n0 = nt << 4;
  const int lo = lane & 15;   // M index for A loads, N index for B loads / D stores
  const int hi = lane >> 4;   // selects K pair (0,1) vs (2,3)

  int arow = m0 + lo;
  if (arow >= M) arow = M - 1;  // clamp: those output rows are discarded anyway
  const float* Xr = X + (long long)arow * K + 2 * hi;
  const float* Wc = W + (long long)(2 * hi) * N + n0 + lo;

  v8f acc = {};
  for (int k = 0; k < K; k += 4) {
    v2f a, b;
    a.x = Xr[0];
    a.y = Xr[1];
    b.x = Wc[0];
    b.y = Wc[N];
    Xr += 4;
    Wc += (long long)4 * N;
    acc = __builtin_amdgcn_wmma_f32_16x16x4_f32(
        /*neg_a=*/false, a, /*neg_b=*/false, b,
        /*c_mod=*/(short)0, acc, /*reuse_a=*/false, /*reuse_b=*/false);
  }

  float* Hp = Hout + (long long)(m0 + hi * 8) * N + n0 + lo;
#pragma unroll
  for (int r = 0; r < 8; ++r) {
    int mr = m0 + hi * 8 + r;
    if (mr < M) Hp[(long long)r * N] = acc[r];
  }
}

// ------------------------------------------------------------------
// Edge scatter: AGG[dst] += H[src] * dinv[src]*dinv[dst]
// one thread = one edge x 4 channels (float4 gather, 4 fp32 global atomics)
// ------------------------------------------------------------------
__device__ __forceinline__ void atomAddF(float* p, float v) {
  unsafeAtomicAdd(p, v);  // lowers to global_atomic_add_f32 on AMD
}

__global__ __launch_bounds__(256) void k_scatter(
    const float* __restrict__ H, const int* __restrict__ src,
    const int* __restrict__ dst, const float* __restrict__ dinv,
    float* __restrict__ AGG, int E) {
  long long tid = (long long)blockIdx.x * blockDim.x + threadIdx.x;
  long long total = (long long)E * (C_HID / 4);
  if (tid >= total) return;
  int e = (int)(tid >> 6);           // C_HID/4 == 64
  int c = ((int)tid & 63) << 2;
  int s = src[e];
  int d = dst[e];
  float nrm = dinv[s] * dinv[d];
  float4 hv = *(const float4*)(H + (long long)s * C_HID + c);
  float* op = AGG + (long long)d * C_HID + c;
  atomAddF(op + 0, hv.x * nrm);
  atomAddF(op + 1, hv.y * nrm);
  atomAddF(op + 2, hv.z * nrm);
  atomAddF(op + 3, hv.w * nrm);
}

// ------------------------------------------------------------------
// OUT = elu(AGG + dinv^2 * H + bias)
// ------------------------------------------------------------------
__global__ __launch_bounds__(256) void k_combine(
    const float* __restrict__ AGG, const float* __restrict__ H,
    const float* __restrict__ dinv, const float* __restrict__ bias,
    float* __restrict__ OUT, int n) {
  long long i = (long long)blockIdx.x * blockDim.x + threadIdx.x;
  long long total = (long long)n * C_HID;
  if (i >= total) return;
  int node = (int)(i >> 8);  // C_HID == 256
  int c = (int)i & 255;
  float di = dinv[node];
  float v = AGG[i] + di * di * H[i] + bias[c];
  OUT[i] = (v > 0.0f) ? v : expm1f(v);
}

// ------------------------------------------------------------------
// Orchestration
// ------------------------------------------------------------------
extern "C" void kernel_launch(void* const* d_in, const int* in_sizes, int n_in,
                              void* d_out, int out_size, void* d_ws, size_t ws_size,
                              hipStream_t stream) {
  const float* x  = (const float*)d_in[0];
  const int*   ei = (const int*)d_in[1];
  const float* W1 = (const float*)d_in[2];
  const float* b1 = (const float*)d_in[3];
  const float* W2 = (const float*)d_in[4];
  const float* b2 = (const float*)d_in[5];
  const float* W3 = (const float*)d_in[6];
  const float* b3 = (const float*)d_in[7];
  float* out = (float*)d_out;

  const int C   = C_HID;
  const int E   = in_sizes[1] / 2;
  const int cin = in_sizes[2] / C;      // 64
  const int n   = in_sizes[0] / cin;    // 10000
  const int* src = ei;
  const int* dst = ei + E;

  // workspace layout (≈31 MB)
  char*  ws  = (char*)d_ws;
  size_t off = 0;
  auto alloc = [&](size_t bytes) -> void* {
    void* p = ws + off;
    off = (off + bytes + 255) & ~(size_t)255;
    return p;
  };
  int*   degi = (int*)  alloc((size_t)n * sizeof(int));
  float* dinv = (float*)alloc((size_t)n * sizeof(float));
  float* H    = (float*)alloc((size_t)n * C * sizeof(float));
  float* AGG  = (float*)alloc((size_t)n * C * sizeof(float));
  float* XB   = (float*)alloc((size_t)n * C * sizeof(float));
  (void)ws_size; (void)n_in; (void)out_size;

  // degrees (once)
  k_deg_init <<<(n + 255) / 256, 256, 0, stream>>>(degi, n);
  k_deg_count<<<(E + 255) / 256, 256, 0, stream>>>(dst, degi, E);
  k_dinv     <<<(n + 255) / 256, 256, 0, stream>>>(degi, dinv, n);

  const long long nC = (long long)n * C;
  const int zgrid = (int)((nC + 255) / 256);
  const long long sThreads = (long long)E * (C / 4);
  const int sgrid = (int)((sThreads + 255) / 256);

  auto layer = [&](const float* Xin, const float* W, const float* bias,
                   float* Out, int K) {
    const int mtiles = (n + 15) / 16;
    const int ntiles = C / 16;
    const int waves  = mtiles * ntiles;
    const int wpb    = 8;  // 256 threads = 8 waves
    const int blocks = (waves + wpb - 1) / wpb;
    k_gemm_wmma<<<blocks, wpb * 32, 0, stream>>>(Xin, W, H, n, K, C);
    k_zero     <<<zgrid, 256, 0, stream>>>(AGG, nC);
    k_scatter  <<<sgrid, 256, 0, stream>>>(H, src, dst, dinv, AGG, E);
    k_combine  <<<zgrid, 256, 0, stream>>>(AGG, H, dinv, bias, Out, n);
  };

  layer(x,  W1, b1, XB,  cin);  // layer 1: K = 64
  layer(XB, W2, b2, XB,  C);    // layer 2: K = 256 (in-place ping is safe: GEMM finishes before combine writes)
  layer(XB, W3, b3, out, C);    // layer 3 -> d_out
}